// UnitGcn_41257455845490
// MI455X (gfx1250) — compile-verified
//
#include <hip/hip_runtime.h>

// Problem constants (from reference)
#define NN  64
#define CC  64
#define TT  300
#define VV  25
#define SS  3
#define II  16
#define TV  (TT*VV)        // 7500
#define CTV (CC*TV)        // 480000

// Stage-1 tiling
#define TB1 12             // timesteps per chunk
#define CH1 (TT/TB1)       // 25 chunks
#define K1  (II*TB1)       // 192 (per-chunk K for scores)

// Stage-2 tiling
#define TB2 6              // timesteps per chunk
#define CH2 (TT/TB2)       // 50 chunks
#define NJ  (TB2*VV)       // 150 columns of Z / y per block

typedef float v2f __attribute__((ext_vector_type(2)));
typedef float v8f __attribute__((ext_vector_type(8)));

static __device__ __forceinline__ v8f wmma_f32(v2f a, v2f b, v8f c) {
  // D = A(16x4, f32) * B(4x16, f32) + C(16x16, f32)
  return __builtin_amdgcn_wmma_f32_16x16x4_f32(false, a, false, b, (short)0, c,
                                               false, false);
}

// ---------------------------------------------------------------------------
// Kernel 1: per-n attention.  scores = (1/(I*T)) * e1 @ e2, softmax over rows,
// adapt = A + W + attn.   One block per n, 8 waves.
// ---------------------------------------------------------------------------
__global__ __launch_bounds__(256)
void gcn_attn_kernel(const float* __restrict__ x,
                     const float* __restrict__ Aadj,
                     const float* __restrict__ Wadj,
                     const float* __restrict__ wa,
                     const float* __restrict__ ba,
                     const float* __restrict__ wb,
                     const float* __restrict__ bb,
                     float* __restrict__ adapt) {
  __shared__ float Als[32 * K1];   // a^T staging: [v(32, pad)][k=i*TB1+t']
  __shared__ float Bls[K1 * 32];   // b   staging: [k][w(32, pad)]
  __shared__ float Sco[32 * 33];   // scores combine buffer

  const int tid  = threadIdx.x;
  const int wave = tid >> 5;
  const int lane = tid & 31;
  const int half = lane >> 4;
  const int lo   = lane & 15;
  const int n    = blockIdx.x;
  const float* xb = x + n * CTV;

  // zero pads once; real entries are rewritten every (chunk, s)
  for (int i = tid; i < 32 * K1; i += 256) Als[i] = 0.f;
  for (int i = tid; i < K1 * 32; i += 256) Bls[i] = 0.f;

  const int vt    = (wave & 3) >> 1;   // scores tile row (v)
  const int wt    = wave & 1;          // scores tile col (w)
  const int khalf = wave >> 2;         // K split among waves

  v8f acc0 = {}, acc1 = {}, acc2 = {};

  for (int ch = 0; ch < CH1; ++ch) {
    const int t0 = ch * TB1;

    // software-prefetch next chunk's x region (64 rows x 300 floats, stride TV)
    if (ch + 1 < CH1) {
      const int nt0 = t0 + TB1;
      for (int idx = tid; idx < 64 * 19; idx += 256) {
        const int c = idx / 19, li = idx % 19;      // 19 cachelines per row
        __builtin_prefetch(xb + c * TV + nt0 * VV + li * 16, 0, 3);
      }
    }

#pragma unroll
    for (int s = 0; s < SS; ++s) {
      __syncthreads();   // previous scores reads of Als/Bls complete
      // ---- embedding GEMMs: a = wa_s @ X, b = wb_s @ X  (M=16,K=64,N=300)
      for (int nt = wave; nt < 19; nt += 8) {            // wave-uniform
        const int j    = lo + 16 * nt;                   // column 0..303
        const int tp   = j / VV;
        const int vcol = j % VV;
        const bool ok  = (j < TB1 * VV);
        const float* wap  = wa + s * (II * CC) + lo * CC;
        const float* wbp  = wb + s * (II * CC) + lo * CC;
        const float* xcol = xb + (t0 + tp) * VV + vcol;
        v8f da = {}, db = {};
#pragma unroll
        for (int kk = 0; kk < 16; ++kk) {
          const int c = kk * 4 + 2 * half;
          v2f bf;
          bf.x = ok ? xcol[c * TV] : 0.f;
          bf.y = ok ? xcol[(c + 1) * TV] : 0.f;
          v2f aw; aw.x = wap[c]; aw.y = wap[c + 1];
          v2f ab; ab.x = wbp[c]; ab.y = wbp[c + 1];
          da = wmma_f32(aw, bf, da);
          db = wmma_f32(ab, bf, db);
        }
        if (ok) {
#pragma unroll
          for (int r = 0; r < 8; ++r) {
            const int i = r + 8 * half;
            const int k = i * TB1 + tp;
            Als[vcol * K1 + k] = da[r] + ba[s * II + i];
            Bls[k * 32 + vcol] = db[r] + bb[s * II + i];
          }
        }
      }
      __syncthreads();
      // ---- scores accumulation: S += a^T @ b  (25x25, K=192 per chunk)
      v8f acc = (s == 0) ? acc0 : (s == 1) ? acc1 : acc2;
      const int arow = (lo + 16 * vt) * K1;
      const int bcol = lo + 16 * wt;
#pragma unroll
      for (int kk = 0; kk < 24; ++kk) {
        const int k0 = (khalf * 24 + kk) * 4 + 2 * half;
        v2f aa; aa.x = Als[arow + k0];      aa.y = Als[arow + k0 + 1];
        v2f bf; bf.x = Bls[k0 * 32 + bcol]; bf.y = Bls[(k0 + 1) * 32 + bcol];
        acc = wmma_f32(aa, bf, acc);
      }
      if (s == 0) acc0 = acc; else if (s == 1) acc1 = acc; else acc2 = acc;
    }
  }

  // ---- combine K-halves, softmax over v (rows) per column w, emit adapt
#pragma unroll
  for (int s = 0; s < SS; ++s) {
    const v8f acc = (s == 0) ? acc0 : (s == 1) ? acc1 : acc2;
    __syncthreads();
    if (khalf == 0) {
#pragma unroll
      for (int r = 0; r < 8; ++r)
        Sco[(r + 8 * half + 16 * vt) * 33 + (lo + 16 * wt)] = acc[r];
    }
    __syncthreads();
    if (khalf == 1) {
#pragma unroll
      for (int r = 0; r < 8; ++r)
        Sco[(r + 8 * half + 16 * vt) * 33 + (lo + 16 * wt)] += acc[r];
    }
    __syncthreads();
    if (tid < VV) {
      const int w = tid;
      float sc[VV];
      float m = -3.4e38f;
#pragma unroll
      for (int v = 0; v < VV; ++v) {
        const float t = Sco[v * 33 + w] * (1.f / (float)(II * TT));
        sc[v] = t;
        m = fmaxf(m, t);
      }
      float sum = 0.f;
#pragma unroll
      for (int v = 0; v < VV; ++v) { sc[v] = expf(sc[v] - m); sum += sc[v]; }
      const float inv  = 1.f / sum;
      const int   base = (n * SS + s) * VV * VV;
#pragma unroll
      for (int v = 0; v < VV; ++v) {
        const int e = v * VV + w;
        adapt[base + e] = Aadj[s * VV * VV + e] + Wadj[s * VV * VV + e]
                        + sc[v] * inv;
      }
    }
  }
}

// ---------------------------------------------------------------------------
// Kernel 2: y[n,o,t,w] = sum_s wd_s @ (X_n @ adapt_{n,s})
// One block per (t-chunk, n).  Also emits deterministic per-block BN partials.
// ---------------------------------------------------------------------------
__global__ __launch_bounds__(256)
void gcn_apply_kernel(const float* __restrict__ x,
                      const float* __restrict__ wd,
                      const float* __restrict__ adapt,
                      float* __restrict__ y,
                      float* __restrict__ part) {
  __shared__ float Zls[CC * NJ];   // Z (then y) staging: [c][j], j=t'*25+w
  __shared__ float Ads[28 * 32];   // adapt tile, K rows padded to 28 (zeros)

  const int tid  = threadIdx.x;
  const int wave = tid >> 5;
  const int lane = tid & 31;
  const int half = lane >> 4;
  const int lo   = lane & 15;
  const int tch  = blockIdx.x;
  const int n    = blockIdx.y;
  const int t0   = tch * TB2;
  const float* xb = x + n * CTV;

  for (int i = tid; i < 28 * 32; i += 256) Ads[i] = 0.f;

  // software-prefetch this block's x region (64 rows x 150 floats, stride TV)
  for (int idx = tid; idx < 64 * 10; idx += 256) {
    const int c = idx / 10, li = idx % 10;          // 10 cachelines per row
    __builtin_prefetch(xb + c * TV + t0 * VV + li * 16, 0, 3);
  }

  v8f yacc[5] = {};   // 40 y-tiles split exactly 5 per wave

#pragma unroll
  for (int s = 0; s < SS; ++s) {
    __syncthreads();   // previous GEMM2 reads of Zls / GEMM1 reads of Ads done
    for (int i = tid; i < VV * VV; i += 256)
      Ads[(i / VV) * 32 + (i % VV)] = adapt[(n * SS + s) * VV * VV + i];
    __syncthreads();

    // ---- GEMM1: Z[(c,t'),w] = sum_v x[c,t0+t',v] * adapt[v,w]  (K=25->28)
    for (int mi = 0; mi < 3; ++mi) {
      const int mt = wave + 8 * mi;           // 24 M-tiles, wave-uniform
      const int m  = lo + 16 * mt;
      const int c  = m / TB2, tp = m % TB2;
      const float* xrow = xb + c * TV + (t0 + tp) * VV;
#pragma unroll
      for (int wt2 = 0; wt2 < 2; ++wt2) {
        v8f dz = {};
#pragma unroll
        for (int kk = 0; kk < 7; ++kk) {
          const int v0 = kk * 4 + 2 * half;
          v2f aa;
          aa.x = (v0     < VV) ? xrow[v0]     : 0.f;
          aa.y = (v0 + 1 < VV) ? xrow[v0 + 1] : 0.f;
          v2f bf;
          bf.x = Ads[v0 * 32 + lo + 16 * wt2];        // pad rows are zero
          bf.y = Ads[(v0 + 1) * 32 + lo + 16 * wt2];
          dz = wmma_f32(aa, bf, dz);
        }
#pragma unroll
        for (int r = 0; r < 8; ++r) {
          const int mm  = 16 * mt + r + 8 * half;
          const int cc2 = mm / TB2, tpp = mm % TB2;
          const int w   = lo + 16 * wt2;
          if (w < VV) Zls[cc2 * NJ + tpp * VV + w] = dz[r];
        }
      }
    }
    __syncthreads();

    // ---- GEMM2: y += wd_s @ Z   (M=64, K=64, N=150 -> 10 N-tiles)
#pragma unroll
    for (int q = 0; q < 5; ++q) {
      const int tt = wave + 8 * q;            // tile 0..39
      const int mt = tt / 10, nt = tt % 10;
      const int j  = lo + 16 * nt;
      const bool ok = (j < NJ);
      const float* wrow = wd + s * CC * CC + (lo + 16 * mt) * CC;
      v8f acc = yacc[q];
#pragma unroll
      for (int kk = 0; kk < 16; ++kk) {
        const int c0 = kk * 4 + 2 * half;
        v2f aa; aa.x = wrow[c0]; aa.y = wrow[c0 + 1];
        v2f bf;
        bf.x = ok ? Zls[c0 * NJ + j]       : 0.f;
        bf.y = ok ? Zls[(c0 + 1) * NJ + j] : 0.f;
        acc = wmma_f32(aa, bf, acc);
      }
      yacc[q] = acc;
    }
  }

  __syncthreads();   // all Zls reads finished before restaging y into it
#pragma unroll
  for (int q = 0; q < 5; ++q) {
    const int tt = wave + 8 * q;
    const int mt = tt / 10, nt = tt % 10;
    const int j  = lo + 16 * nt;
    if (j < NJ) {
      const int tp = j / VV, w = j % VV;
#pragma unroll
      for (int r = 0; r < 8; ++r) {
        const int o = 16 * mt + r + 8 * half;
        const float v = yacc[q][r];
        y[n * CTV + o * TV + (t0 + tp) * VV + w] = v;
        Zls[o * NJ + j] = v;
      }
    }
  }
  __syncthreads();

  // deterministic per-block BN partial sums (fixed order, no atomics)
  if (tid < CC) {
    float s1 = 0.f, s2 = 0.f;
    for (int j = 0; j < NJ; ++j) {
      const float v = Zls[tid * NJ + j];
      s1 += v; s2 += v * v;
    }
    const int blk = n * CH2 + tch;
    part[blk * 128 + tid]      = s1;
    part[blk * 128 + 64 + tid] = s2;
  }
}

// ---------------------------------------------------------------------------
// Kernel 3: reduce BN partials (fixed order) -> per-channel scale/shift.
// bd is a per-channel constant and cancels exactly inside batch-norm.
// ---------------------------------------------------------------------------
__global__ void gcn_bnstats_kernel(const float* __restrict__ part,
                                   const float* __restrict__ gamma,
                                   const float* __restrict__ beta,
                                   float* __restrict__ ss) {
  const int ch = threadIdx.x;
  if (ch < CC) {
    float s1 = 0.f, s2 = 0.f;
    for (int b = 0; b < NN * CH2; ++b) {
      s1 += part[b * 128 + ch];
      s2 += part[b * 128 + 64 + ch];
    }
    const float cnt  = (float)(NN * TT * VV);
    const float mean = s1 / cnt;
    const float var  = s2 / cnt - mean * mean;
    const float sc   = gamma[ch] * rsqrtf(var + 1e-5f);
    ss[ch]      = sc;
    ss[64 + ch] = beta[ch] - mean * sc;
  }
}

// ---------------------------------------------------------------------------
// Kernel 4: out = relu(y * scale + shift + x), float4-vectorized.
// 7500 floats per (n,c) row = 1875 float4 -> vectors never cross channels.
// ---------------------------------------------------------------------------
__global__ __launch_bounds__(256)
void gcn_bnrelu_kernel(const float* __restrict__ y,
                       const float* __restrict__ x,
                       const float* __restrict__ ss,
                       float* __restrict__ out) {
  const int idx = blockIdx.x * 256 + threadIdx.x;   // float4 index
  const int o   = (idx / (TV / 4)) & (CC - 1);
  const float4 yv = ((const float4*)y)[idx];
  const float4 xv = ((const float4*)x)[idx];
  const float sc = ss[o], sh = ss[64 + o];
  float4 r;
  r.x = fmaxf(yv.x * sc + sh + xv.x, 0.f);
  r.y = fmaxf(yv.y * sc + sh + xv.y, 0.f);
  r.z = fmaxf(yv.z * sc + sh + xv.z, 0.f);
  r.w = fmaxf(yv.w * sc + sh + xv.w, 0.f);
  ((float4*)out)[idx] = r;
}

// ---------------------------------------------------------------------------
extern "C" void kernel_launch(void* const* d_in, const int* in_sizes, int n_in,
                              void* d_out, int out_size, void* d_ws,
                              size_t ws_size, hipStream_t stream) {
  const float* x     = (const float*)d_in[0];
  const float* Aadj  = (const float*)d_in[1];
  const float* Wadj  = (const float*)d_in[2];
  const float* wa    = (const float*)d_in[3];
  const float* ba    = (const float*)d_in[4];
  const float* wb    = (const float*)d_in[5];
  const float* bb    = (const float*)d_in[6];
  const float* wd    = (const float*)d_in[7];
  // d_in[8] = bd: per-channel constant, cancels inside batch-norm -> unused
  const float* gamma = (const float*)d_in[9];
  const float* beta  = (const float*)d_in[10];
  float* out = (float*)d_out;

  float* ws    = (float*)d_ws;
  float* adapt = ws;                                   // 192*625
  float* yb    = adapt + NN * SS * VV * VV;            // 30,720,000
  float* part  = yb + NN * CC * TT * VV;               // 3200*128
  float* ss    = part + NN * CH2 * 128;                // 128

  gcn_attn_kernel<<<NN, 256, 0, stream>>>(x, Aadj, Wadj, wa, ba, wb, bb, adapt);
  gcn_apply_kernel<<<dim3(CH2, NN), 256, 0, stream>>>(x, wd, adapt, yb, part);
  gcn_bnstats_kernel<<<1, 64, 0, stream>>>(part, gamma, beta, ss);
  gcn_bnrelu_kernel<<<(NN * CC * TT * VV) / 4 / 256, 256, 0, stream>>>(yb, x, ss,
                                                                       out);
}